// InvariantPointAttention_48533130444939
// MI455X (gfx1250) — compile-verified
//
#include <hip/hip_runtime.h>
#include <hip/hip_bf16.h>
#include <math.h>

typedef __attribute__((ext_vector_type(2))) float    v2f;
typedef __attribute__((ext_vector_type(8))) float    v8f;
typedef __attribute__((ext_vector_type(4))) unsigned uint32x4;
typedef __attribute__((ext_vector_type(8))) int      int32x8;
typedef __attribute__((ext_vector_type(4))) int      int32x4;

#define SCALE_QK 1.63299316185545206546f   /* sqrt(8/3)   */
#define SCALE_B  0.57735026918962576451f   /* sqrt(1/3)   */
#define SCALE_V  0.09622504486493762742f   /* sqrt(1/108) */

// ---------- sortable-uint mapping for float atomic max ----------
__device__ __forceinline__ unsigned fkey(float f) {
  unsigned u = __float_as_uint(f);
  return (u & 0x80000000u) ? ~u : (u | 0x80000000u);
}
__device__ __forceinline__ float funkey(unsigned k) {
  unsigned u = (k & 0x80000000u) ? (k & 0x7FFFFFFFu) : ~k;
  return __uint_as_float(u);
}

// ---------- generic fp32 WMMA GEMM: C(MxP) = A(MxK) @ W(KxP) [+ bias] ----------
__global__ void gemm_wmma_f32(const float* __restrict__ A, const float* __restrict__ W,
                              const float* __restrict__ bias, float* __restrict__ C,
                              int M, int K, int P) {
  int wave = (int)((blockIdx.x * blockDim.x + threadIdx.x) >> 5);
  int lane = threadIdx.x & 31;
  int pt = P >> 4;
  int ntiles = (M >> 4) * pt;
  if (wave >= ntiles) return;              // uniform per wave: EXEC stays all-ones
  int mt = wave / pt, nt = wave - mt * pt;
  int l = lane & 15, half = lane >> 4;

  const float* arow = A + (size_t)(mt * 16 + l) * K + 2 * half; // A frag: lanes 0-15 K0/K1, 16-31 K2/K3
  const float* wcol = W + (size_t)(2 * half) * P + nt * 16 + l; // B frag mirrors A layout

  v8f acc = {};
  for (int k = 0; k < K; k += 4) {
    v2f a = *(const v2f*)(arow + k);
    v2f b;
    b.x = wcol[(size_t)k * P];
    b.y = wcol[(size_t)(k + 1) * P];
    acc = __builtin_amdgcn_wmma_f32_16x16x4_f32(false, a, false, b, (short)0, acc, false, false);
  }
  float bv = bias ? bias[nt * 16 + l] : 0.0f;
  #pragma unroll
  for (int r = 0; r < 8; ++r) {
    int row = mt * 16 + r + 8 * half;      // C/D layout: lane half selects M+8
    C[(size_t)row * P + nt * 16 + l] = acc[r] + bv;
  }
}

// ---------- build padded combined edge weight: 64x32 = [Wb(64x4) | Wdz(64x16) | 0] ----------
__global__ void build_wcomb(const float* __restrict__ Wb, const float* __restrict__ Wdz,
                            float* __restrict__ Wcomb) {
  int t = blockIdx.x * blockDim.x + threadIdx.x;
  if (t >= 64 * 32) return;
  int krow = t >> 5, col = t & 31;
  float v = 0.0f;
  if (col < 4)       v = Wb[krow * 4 + col];
  else if (col < 20) v = Wdz[krow * 16 + (col - 4)];
  Wcomb[t] = v;
}

// ---------- TDM issue: DMA one 128x64 fp32 tile of z (contiguous 32 KB) into LDS ----------
// D# per CDNA5 ISA ch.8: group0 = {count, lds_addr, global_addr, type=2};
// group1 = {data_size=4B, tensor_dim0=64, tensor_dim1=128, tile_dim0=64, tile_dim1=128, stride0=64}.
__device__ __forceinline__ void tdm_load_z_tile(const float* z, int tile, unsigned lds_off) {
  unsigned long long g = (unsigned long long)(uintptr_t)z
                       + (unsigned long long)tile * (128ull * 64ull * 4ull);
  uint32x4 g0;
  g0.x = 1u;                                                     // count=1 (valid user descriptor)
  g0.y = lds_off;                                                // LDS byte address of tile
  g0.z = (unsigned)(g & 0xFFFFFFFFull);                          // global_addr[31:0]
  g0.w = (unsigned)((g >> 32) & 0x01FFFFFFull) | (2u << 30);     // global_addr[56:32] | type=2
  int32x8 g1;
  g1[0] = (2 << 16);     // data_size = 2 -> 4 bytes; no multicast/iterate/pad
  g1[1] = (64 << 16);    // tensor_dim0[15:0] = 64 (row length in elements)
  g1[2] = (128 << 16);   // tensor_dim1[15:0] = 128 (rows in this tile)
  g1[3] = (64 << 16);    // tile_dim0 = 64
  g1[4] = 128;           // tile_dim1 = 128, tile_dim2 = 0
  g1[5] = 64;            // tensor_dim0_stride[31:0] = 64 elements
  g1[6] = 0;             // stride0[47:32]=0, tensor_dim1_stride[15:0]=0
  g1[7] = 0;
  int32x4 zr4 = {0, 0, 0, 0};                                    // groups 2/3 unused (2-D tensor)
  int32x8 zr8 = {0, 0, 0, 0, 0, 0, 0, 0};
  __builtin_amdgcn_tensor_load_to_lds(g0, g1, zr4, zr4, zr8, 0); // 6-arg toolchain form
}

// ---------- z @ Wcomb: TDM double-buffered LDS staging + WMMA, split-write b_bias/pair ----------
__global__ void zproj_tdm(const float* __restrict__ z, const float* __restrict__ Wcomb,
                          const float* __restrict__ bb, const float* __restrict__ bdz,
                          float* __restrict__ bbias, float* __restrict__ pair, int E) {
  __shared__ float tile[2][128 * 64];                            // 2 x 32 KB double buffer
  const int T = E >> 7;                                          // tiles of 128 edge rows
  int wave = threadIdx.x >> 5;
  int lane = threadIdx.x & 31;
  int l = lane & 15, half = lane >> 4;
  int G = (int)gridDim.x;

  if (wave == 0 && (int)blockIdx.x < T)                          // prime the pipeline
    tdm_load_z_tile(z, blockIdx.x, (unsigned)(uintptr_t)&tile[0][0]);

  int cur = 0;
  for (int t = blockIdx.x; t < T; t += G) {
    int tn = t + G;
    if (wave == 0) {                                             // wave-uniform: one DMA per block
      if (tn < T) {
        tdm_load_z_tile(z, tn, (unsigned)(uintptr_t)&tile[cur ^ 1][0]);
        __builtin_amdgcn_s_wait_tensorcnt(1);                    // tile t done, t+1 in flight
      } else {
        __builtin_amdgcn_s_wait_tensorcnt(0);
      }
    }
    __syncthreads();                                             // publish LDS tile to all 8 waves

    // each wave: 16 rows x 32 cols via two fp32 WMMA accumulators, A from LDS (ds_load_b64)
    const float* asrc = &tile[cur][(wave * 16 + l) * 64 + 2 * half];
    #pragma unroll
    for (int nt = 0; nt < 2; ++nt) {
      v8f acc = {};
      const float* wc = Wcomb + (size_t)(2 * half) * 32 + nt * 16 + l;
      #pragma unroll
      for (int k = 0; k < 64; k += 4) {
        v2f a = *(const v2f*)(asrc + k);
        v2f b;
        b.x = wc[k * 32];
        b.y = wc[(k + 1) * 32];
        acc = __builtin_amdgcn_wmma_f32_16x16x4_f32(false, a, false, b, (short)0, acc, false, false);
      }
      int col = nt * 16 + l;
      #pragma unroll
      for (int r = 0; r < 8; ++r) {
        int row = t * 128 + wave * 16 + r + 8 * half;
        float v = acc[r];
        if (col < 4)       bbias[(size_t)row * 4 + col]       = v + bb[col];
        else if (col < 20) pair[(size_t)row * 16 + (col - 4)] = v + bdz[col - 4];
      }
    }
    __syncthreads();                                             // protect buffer before next DMA
    cur ^= 1;
  }
}

// ---------- quat rotate + translate raw projections (N,3,32) -> (N,32,3) ----------
__global__ void rotate_points(const float* __restrict__ raw, const float* __restrict__ quat,
                              const float* __restrict__ trans, float* __restrict__ out, int n) {
  int t = blockIdx.x * blockDim.x + threadIdx.x;
  int node = t >> 5, p = t & 31;
  if (node >= n) return;
  float vx = raw[(size_t)node * 96 + p];
  float vy = raw[(size_t)node * 96 + 32 + p];
  float vz = raw[(size_t)node * 96 + 64 + p];
  float w  = quat[node * 4 + 0], ux = quat[node * 4 + 1];
  float uy = quat[node * 4 + 2], uz = quat[node * 4 + 3];
  float cx = uy * vz - uz * vy, cy = uz * vx - ux * vz, cz = ux * vy - uy * vx;
  float rx = vx + 2.0f * (w * cx + (uy * cz - uz * cy));
  float ry = vy + 2.0f * (w * cy + (uz * cx - ux * cz));
  float rz = vz + 2.0f * (w * cz + (ux * cy - uy * cx));
  out[(size_t)node * 96 + p * 3 + 0] = rx + trans[node * 3 + 0];
  out[(size_t)node * 96 + p * 3 + 1] = ry + trans[node * 3 + 1];
  out[(size_t)node * 96 + p * 3 + 2] = rz + trans[node * 3 + 2];
}

// ---------- init: denom=0, segmax=key(-inf), outpair=0 ----------
__global__ void init_buffers(float* __restrict__ denom, unsigned* __restrict__ segmax,
                             float* __restrict__ outpair, int n) {
  int t = blockIdx.x * blockDim.x + threadIdx.x;
  if (t < n * 4) { denom[t] = 0.0f; segmax[t] = 0x007FFFFFu; }
  if (t < n * 64) outpair[t] = 0.0f;
}

// ---------- per (edge,head) attention logit + running segment max ----------
__global__ void edge_logits(const float* __restrict__ qs, const float* __restrict__ ks,
                            const float* __restrict__ qvrt, const float* __restrict__ kvrt,
                            const float* __restrict__ bbias, const int* __restrict__ ei,
                            const float* __restrict__ hwraw, float* __restrict__ attn,
                            unsigned* __restrict__ segmax, int E) {
  int t = blockIdx.x * blockDim.x + threadIdx.x;
  if (t >= E * 4) return;
  int e = t >> 2, h = t & 3;
  int i = ei[e], j = ei[E + e];
  const float4* a = (const float4*)(ks + (size_t)i * 256 + h * 64);
  const float4* b = (const float4*)(qs + (size_t)j * 256 + h * 64);
  float dot = 0.0f;
  #pragma unroll
  for (int u = 0; u < 16; ++u) {
    float4 x = a[u], y = b[u];
    dot += x.x * y.x + x.y * y.y + x.z * y.z + x.w * y.w;
  }
  const float* pq = qvrt + (size_t)i * 96 + h * 24;
  const float* pk = kvrt + (size_t)j * 96 + h * 24;
  float ss = 0.0f;
  #pragma unroll
  for (int c = 0; c < 24; ++c) { float d = pq[c] - pk[c]; ss += d * d; }
  float hw = log1pf(expf(hwraw[h]));  // softplus
  float att = dot * SCALE_QK + SCALE_B * bbias[(size_t)e * 4 + h]
            - 0.5f * SCALE_V * hw * ss;
  attn[t] = att;
  atomicMax(segmax + (size_t)i * 4 + h, fkey(att));
}

// ---------- exp(attn - segmax) in-place + denom accumulation ----------
__global__ void edge_exp(float* __restrict__ attn, const unsigned* __restrict__ segmax,
                         float* __restrict__ denom, const int* __restrict__ ei, int E) {
  int t = blockIdx.x * blockDim.x + threadIdx.x;
  if (t >= E * 4) return;
  int e = t >> 2, h = t & 3;
  int i = ei[e];
  float m = funkey(segmax[(size_t)i * 4 + h]);
  float ea = expf(attn[t] - m);
  attn[t] = ea;
  atomicAdd(denom + (size_t)i * 4 + h, ea);
}

// ---------- normalized attention scatter (out_pair is the ONLY true segment sum) ----------
__global__ void edge_scatter_pair(const float* __restrict__ attn, const float* __restrict__ denom,
                                  const float* __restrict__ pair, const int* __restrict__ ei,
                                  float* __restrict__ outpair, int E) {
  int t = blockIdx.x * blockDim.x + threadIdx.x;
  if (t >= E * 4) return;
  int e = t >> 2, h = t & 3;
  int i = ei[e];
  float w = attn[t] / denom[(size_t)i * 4 + h];
  const float* pz = pair + (size_t)e * 16;
  float* dst = outpair + (size_t)i * 64 + h * 16;
  #pragma unroll
  for (int c = 0; c < 16; ++c) atomicAdd(dst + c, w * pz[c]);
}

// ---------- per-node feature assembly: out_scalar==v_scalar, out_vec==v_vec (softmax sums to 1) ----------
__global__ void assemble_feats(const float* __restrict__ vs, const float* __restrict__ vvrt,
                               const float* __restrict__ quat, const float* __restrict__ trans,
                               const float* __restrict__ denom, const float* __restrict__ outpair,
                               float* __restrict__ feats, int n) {
  int node = blockIdx.x;
  if (node >= n) return;
  int tid = threadIdx.x;
  float has = (denom[(size_t)node * 4] > 0.0f) ? 1.0f : 0.0f;  // empty-segment mask
  float* f = feats + (size_t)node * 448;
  const float* v = vs + (size_t)node * 256;
  for (int k = tid; k < 256; k += 64) f[k] = v[k] * has;
  if (tid < 32) {
    int p = tid;
    const float* pt = vvrt + (size_t)node * 96 + p * 3;
    float tx = trans[node * 3 + 0], ty = trans[node * 3 + 1], tz = trans[node * 3 + 2];
    float vx = pt[0] * has - tx, vy = pt[1] * has - ty, vz = pt[2] * has - tz;
    float w  = quat[node * 4 + 0];
    float ux = -quat[node * 4 + 1], uy = -quat[node * 4 + 2], uz = -quat[node * 4 + 3]; // conj
    float cx = uy * vz - uz * vy, cy = uz * vx - ux * vz, cz = ux * vy - uy * vx;
    float rx = vx + 2.0f * (w * cx + (uy * cz - uz * cy));
    float ry = vy + 2.0f * (w * cy + (uz * cx - ux * cz));
    float rz = vz + 2.0f * (w * cz + (ux * cy - uy * cx));
    f[256 + p]      = rx;
    f[256 + 32 + p] = ry;
    f[256 + 64 + p] = rz;
    f[256 + 96 + p] = sqrtf(rx * rx + ry * ry + rz * rz + 1e-8f);
  }
  const float* op = outpair + (size_t)node * 64;
  for (int k = tid; k < 64; k += 64) f[352 + k] = op[k];
}

static inline size_t alignup(size_t x) { return (x + 255) & ~(size_t)255; }

extern "C" void kernel_launch(void* const* d_in, const int* in_sizes, int n_in,
                              void* d_out, int out_size, void* d_ws, size_t ws_size,
                              hipStream_t stream) {
  const float* s      = (const float*)d_in[0];
  const float* z      = (const float*)d_in[1];
  const int*   ei     = (const int*)  d_in[2];
  const float* quat   = (const float*)d_in[3];
  const float* trans  = (const float*)d_in[4];
  const float* Wq_s   = (const float*)d_in[5];
  const float* Wk_s   = (const float*)d_in[6];
  const float* Wv_s   = (const float*)d_in[7];
  const float* Wq_v   = (const float*)d_in[8];
  const float* Wk_v   = (const float*)d_in[9];
  const float* Wv_v   = (const float*)d_in[10];
  const float* Wb     = (const float*)d_in[11];
  const float* bb     = (const float*)d_in[12];
  const float* Wdz    = (const float*)d_in[13];
  const float* bdz    = (const float*)d_in[14];
  const float* hwraw  = (const float*)d_in[15];
  const float* Wout   = (const float*)d_in[16];
  const float* bout   = (const float*)d_in[17];
  float* out = (float*)d_out;

  const int n = in_sizes[0] / 64;   // 20000
  const int E = in_sizes[1] / 64;   // 640000

  // ---- workspace layout ----
  char* p = (char*)d_ws;
  size_t off = 0;
  auto take = [&](size_t bytes) { char* r = p + off; off = alignup(off + bytes); return (void*)r; };
  float*    qs      = (float*)   take((size_t)n * 256 * 4);
  float*    ks      = (float*)   take((size_t)n * 256 * 4);
  float*    vs      = (float*)   take((size_t)n * 256 * 4);
  float*    rawv    = (float*)   take((size_t)n * 96 * 4);   // reused for q/k/v via stream order
  float*    qvrt    = (float*)   take((size_t)n * 96 * 4);
  float*    kvrt    = (float*)   take((size_t)n * 96 * 4);
  float*    vvrt    = (float*)   take((size_t)n * 96 * 4);
  float*    Wcomb   = (float*)   take(64 * 32 * 4);
  float*    bbias   = (float*)   take((size_t)E * 4 * 4);
  float*    pair    = (float*)   take((size_t)E * 16 * 4);
  float*    attn    = (float*)   take((size_t)E * 4 * 4);
  unsigned* segmax  = (unsigned*)take((size_t)n * 4 * 4);
  float*    denom   = (float*)   take((size_t)n * 4 * 4);
  float*    outpair = (float*)   take((size_t)n * 64 * 4);
  float*    feats   = (float*)   take((size_t)n * 448 * 4);
  (void)ws_size; (void)n_in; (void)out_size;

  // ---- init accumulators ----
  init_buffers<<<(n * 64 + 255) / 256, 256, 0, stream>>>(denom, segmax, outpair, n);
  build_wcomb<<<(64 * 32 + 255) / 256, 256, 0, stream>>>(Wb, Wdz, Wcomb);

  // ---- node projections (WMMA fp32) ----
  {
    int waves = (n / 16) * (256 / 16);
    gemm_wmma_f32<<<(waves + 7) / 8, 256, 0, stream>>>(s, Wq_s, nullptr, qs, n, 64, 256);
    gemm_wmma_f32<<<(waves + 7) / 8, 256, 0, stream>>>(s, Wk_s, nullptr, ks, n, 64, 256);
    gemm_wmma_f32<<<(waves + 7) / 8, 256, 0, stream>>>(s, Wv_s, nullptr, vs, n, 64, 256);
  }
  {
    int waves = (n / 16) * (96 / 16);
    int rblk = (n * 32 + 255) / 256;
    gemm_wmma_f32<<<(waves + 7) / 8, 256, 0, stream>>>(s, Wq_v, nullptr, rawv, n, 64, 96);
    rotate_points<<<rblk, 256, 0, stream>>>(rawv, quat, trans, qvrt, n);
    gemm_wmma_f32<<<(waves + 7) / 8, 256, 0, stream>>>(s, Wk_v, nullptr, rawv, n, 64, 96);
    rotate_points<<<rblk, 256, 0, stream>>>(rawv, quat, trans, kvrt, n);
    gemm_wmma_f32<<<(waves + 7) / 8, 256, 0, stream>>>(s, Wv_v, nullptr, rawv, n, 64, 96);
    rotate_points<<<rblk, 256, 0, stream>>>(rawv, quat, trans, vvrt, n);
  }

  // ---- single pass over z: TDM-staged fused b_bias + pair_z (tensor_load_to_lds + WMMA) ----
  {
    int tiles = E / 128;
    int blocks = tiles < 500 ? tiles : 500;   // grid-stride, double-buffered DMA per block
    zproj_tdm<<<blocks, 256, 0, stream>>>(z, Wcomb, bb, bdz, bbias, pair, E);
  }

  // ---- edge attention: logits -> segment max -> exp/denom -> pair scatter ----
  {
    int thr = E * 4, blk = (thr + 255) / 256;
    edge_logits<<<blk, 256, 0, stream>>>(qs, ks, qvrt, kvrt, bbias, ei, hwraw, attn, segmax, E);
    edge_exp<<<blk, 256, 0, stream>>>(attn, segmax, denom, ei, E);
    edge_scatter_pair<<<blk, 256, 0, stream>>>(attn, denom, pair, ei, outpair, E);
  }

  // ---- assemble features + final WMMA GEMM ----
  assemble_feats<<<n, 64, 0, stream>>>(vs, vvrt, quat, trans, denom, outpair, feats, n);
  {
    int waves = (n / 16) * (64 / 16);
    gemm_wmma_f32<<<(waves + 7) / 8, 256, 0, stream>>>(feats, Wout, bout, out, n, 448, 64);
  }
}